// PhaseLoss_26448408608917
// MI455X (gfx1250) — compile-verified
//
#include <hip/hip_runtime.h>
#include <hip/hip_bf16.h>

// PhaseLoss: B=32, F=513, T=1600  (fixed by the reference)
#define PL_B   32
#define PL_F   513
#define PL_T   1600
#define PL_V   (PL_T / 4)          // 400 float4 per row
#define PL_ROWS (PL_B * PL_F)      // 16416
#define PL_BLOCKS (PL_ROWS / 8)    // 2052 blocks, 8 rows each
#define PL_N   ((float)PL_B * (float)PL_F * (float)PL_T) // 26265600

typedef __attribute__((ext_vector_type(2))) float v2f;
typedef __attribute__((ext_vector_type(8))) float v8f;

__device__ __forceinline__ float pl_anti_wrap(float x) {
    const float TWO_PI     = 6.28318530717958647692f;
    const float INV_TWO_PI = 0.15915494309189533577f;
    float n = rintf(x * INV_TWO_PI);   // round-half-even, matches jnp.round
    return fabsf(x - n * TWO_PI);
}

// Sum a value across all 32 lanes of a wave. Result valid in every lane.
// Uses V_WMMA_F32_16X16X4_F32 with B = ones: D[m][n] = sum_k A[m][k]
// (A holds one partial per lane in K-slot 0, zero in slot 1), so
// sum over D VGPRs 0..7 leaves half-sums in lane groups 0-15 / 16-31,
// finished with one xor-16 exchange.
__device__ __forceinline__ float pl_wave_sum(float v) {
#if defined(__gfx1250__) && __has_builtin(__builtin_amdgcn_wmma_f32_16x16x4_f32)
    v2f a; a.x = v;    a.y = 0.0f;
    v2f b; b.x = 1.0f; b.y = 1.0f;
    v8f c = {};
    v8f d = __builtin_amdgcn_wmma_f32_16x16x4_f32(
        /*neg_a=*/false, a, /*neg_b=*/false, b,
        /*c_mod=*/(short)0, c, /*reuse_a=*/false, /*reuse_b=*/false);
    float s = ((d[0] + d[1]) + (d[2] + d[3])) + ((d[4] + d[5]) + (d[6] + d[7]));
    s += __shfl_xor(s, 16, 32);
    return s;
#else
    for (int m = 16; m >= 1; m >>= 1) v += __shfl_xor(v, m, 32);
    return v;
#endif
}

__global__ __launch_bounds__(256) void phase_loss_main(
    const float* __restrict__ pr, const float* __restrict__ pg,
    float* __restrict__ part) {

    __shared__ float4 lds4[PL_V];       // current row's d values
    __shared__ float  wsum[8][3];       // per-wave partials

    const int tid  = threadIdx.x;
    const int lane = tid & 31;
    const int wave = tid >> 5;

    float s_ip = 0.0f, s_gd = 0.0f, s_ptd = 0.0f;

    const int row0 = blockIdx.x * 8;

    for (int rr = 0; rr < 8; ++rr) {
        const int row = row0 + rr;
        const int f   = row % PL_F;
        const bool hasPrev = (f != 0);

        const float4* rc = (const float4*)(pr + (size_t)row * PL_T);
        const float4* gc = (const float4*)(pg + (size_t)row * PL_T);
        const float4* rp = (const float4*)(pr + (size_t)(row - 1) * PL_T);
        const float4* gp = (const float4*)(pg + (size_t)(row - 1) * PL_T);

        // Pass 1: stream both rows, build d in LDS, accumulate ip/gd and
        // the intra-float4 ptd terms.
        for (int v = tid; v < PL_V; v += 256) {
            // prefetch next row (contiguous layout) -> global_prefetch_b8
            __builtin_prefetch(&rc[v + PL_V], 0, 1);
            __builtin_prefetch(&gc[v + PL_V], 0, 1);

            float4 a = rc[v];
            float4 b = gc[v];
            float4 d;
            d.x = a.x - b.x; d.y = a.y - b.y;
            d.z = a.z - b.z; d.w = a.w - b.w;
            lds4[v] = d;

            float tip = pl_anti_wrap(d.x) + pl_anti_wrap(d.y) +
                        pl_anti_wrap(d.z) + pl_anti_wrap(d.w);
            s_ip += tip;

            if (hasPrev) {
                float4 ap = rp[v];   // loaded by this block last iteration -> cache hit
                float4 bp = gp[v];
                s_gd += pl_anti_wrap((ap.x - bp.x) - d.x)
                      + pl_anti_wrap((ap.y - bp.y) - d.y)
                      + pl_anti_wrap((ap.z - bp.z) - d.z)
                      + pl_anti_wrap((ap.w - bp.w) - d.w);
            } else {
                // anti_wrap(0 - d) == anti_wrap(d)
                s_gd += tip;
            }

            // ptd terms fully inside this float4 (t = 4v+1..4v+3)
            s_ptd += pl_anti_wrap(d.x - d.y)
                   + pl_anti_wrap(d.y - d.z)
                   + pl_anti_wrap(d.z - d.w);
        }
        __syncthreads();

        // Pass 2: ptd boundary terms at t = 4v (need d[4v-1] from LDS).
        for (int v = tid; v < PL_V; v += 256) {
            float dm1 = (v == 0) ? 0.0f : lds4[v - 1].w;
            float dx  = lds4[v].x;
            s_ptd += pl_anti_wrap(dm1 - dx);
        }
        __syncthreads();  // before next row overwrites lds4
    }

    // Block reduction: wave32 WMMA reduce, then 8 wave partials via LDS.
    float w_ip  = pl_wave_sum(s_ip);
    float w_gd  = pl_wave_sum(s_gd);
    float w_ptd = pl_wave_sum(s_ptd);
    if (lane == 0) {
        wsum[wave][0] = w_ip;
        wsum[wave][1] = w_gd;
        wsum[wave][2] = w_ptd;
    }
    __syncthreads();
    if (tid == 0) {
        float t0 = 0.f, t1 = 0.f, t2 = 0.f;
        for (int w = 0; w < 8; ++w) {
            t0 += wsum[w][0]; t1 += wsum[w][1]; t2 += wsum[w][2];
        }
        part[0 * PL_BLOCKS + blockIdx.x] = t0;
        part[1 * PL_BLOCKS + blockIdx.x] = t1;
        part[2 * PL_BLOCKS + blockIdx.x] = t2;
    }
}

__global__ __launch_bounds__(256) void phase_loss_finalize(
    const float* __restrict__ part, float* __restrict__ out) {

    __shared__ float wsum[8];
    const int tid  = threadIdx.x;
    const int lane = tid & 31;
    const int wave = tid >> 5;
    const float invN = 1.0f / PL_N;

    for (int l = 0; l < 3; ++l) {
        float s = 0.0f;
        for (int i = tid; i < PL_BLOCKS; i += 256)
            s += part[(size_t)l * PL_BLOCKS + i];
        s = pl_wave_sum(s);
        if (lane == 0) wsum[wave] = s;
        __syncthreads();
        if (tid == 0) {
            float t = 0.f;
            for (int w = 0; w < 8; ++w) t += wsum[w];
            out[l] = t * invN;
        }
        __syncthreads();
    }
}

extern "C" void kernel_launch(void* const* d_in, const int* in_sizes, int n_in,
                              void* d_out, int out_size, void* d_ws, size_t ws_size,
                              hipStream_t stream) {
    const float* pr = (const float*)d_in[0];   // phase_r, f32 (32,513,1600)
    const float* pg = (const float*)d_in[1];   // phase_g, f32 (32,513,1600)
    float* part = (float*)d_ws;                // 3 * 2052 floats of scratch
    float* out  = (float*)d_out;               // 3 scalars (ip, gd, ptd)

    phase_loss_main<<<PL_BLOCKS, 256, 0, stream>>>(pr, pg, part);
    phase_loss_finalize<<<1, 256, 0, stream>>>(part, out);
}